// RNNModel_1529008357793
// MI455X (gfx1250) — compile-verified
//
#include <hip/hip_runtime.h>
#include <hip/hip_bf16.h>

#define HID 1000
#define VOC 32000

typedef float v2f __attribute__((ext_vector_type(2)));
typedef float v8f __attribute__((ext_vector_type(8)));

// ---------------------------------------------------------------------------
// Device-scope sense-reversing grid barrier (125 resident blocks, trivially
// co-resident on MI455X). count must be 0 at launch (memset by kernel_launch).
// ---------------------------------------------------------------------------
__device__ __forceinline__ void grid_barrier(unsigned* count, unsigned* gen,
                                             unsigned nblocks) {
    __syncthreads();
    if (threadIdx.x == 0) {
        __threadfence();  // publish this block's h writes at agent scope
        unsigned my_gen = __hip_atomic_load(gen, __ATOMIC_ACQUIRE,
                                            __HIP_MEMORY_SCOPE_AGENT);
        unsigned prev = __hip_atomic_fetch_add(count, 1u, __ATOMIC_ACQ_REL,
                                               __HIP_MEMORY_SCOPE_AGENT);
        if (prev == nblocks - 1u) {
            __hip_atomic_store(count, 0u, __ATOMIC_RELAXED,
                               __HIP_MEMORY_SCOPE_AGENT);
            __hip_atomic_store(gen, my_gen + 1u, __ATOMIC_RELEASE,
                               __HIP_MEMORY_SCOPE_AGENT);
        } else {
            while (__hip_atomic_load(gen, __ATOMIC_ACQUIRE,
                                     __HIP_MEMORY_SCOPE_AGENT) == my_gen) {
                __builtin_amdgcn_s_sleep(2);
            }
        }
        __threadfence();
    }
    __syncthreads();
}

// ---------------------------------------------------------------------------
// Kernel A: sequential RNN scan. 125 blocks x 256 threads (8 waves).
// Wave w of block b owns hidden row r = b*8+w; its W_hh row lives in LDS.
// Per step: dot(wrow, h_t) via 32 lanes + shfl reduction, + gathered x, tanh.
// all_h rows written to workspace; grid barrier between steps.
// ---------------------------------------------------------------------------
__global__ __launch_bounds__(256) void rnn_scan_kernel(
    const int* __restrict__ tok, const float* __restrict__ h0,
    const float* __restrict__ Wxh, const float* __restrict__ Whh,
    const float* __restrict__ Bh, float* __restrict__ allH,
    unsigned* __restrict__ barCount, unsigned* __restrict__ barGen,
    int T, unsigned nblocks) {
    __shared__ float wrow[8 * HID];  // 32 KB: 8 rows of W_hh

    const int lane = threadIdx.x & 31;
    const int wave = threadIdx.x >> 5;
    const int r = blockIdx.x * 8 + wave;  // 0..999 exactly (125*8 = 1000)

    // Stage this wave's W_hh row into LDS (stays resident for all T steps).
    for (int k = lane; k < HID; k += 32)
        wrow[wave * HID + k] = Whh[(size_t)r * HID + k];

    // all_h[0] = initial_h
    if (lane == 0) allH[r] = h0[r];

    grid_barrier(barCount, barGen, nblocks);

    const float bh = Bh[r];
    const float* wxr = Wxh + (size_t)r * VOC;

    for (int t = 0; t < T; ++t) {
        const float* h = allH + (size_t)t * HID;
        float sum = 0.f;
        for (int k = lane; k < HID; k += 32)
            sum = fmaf(wrow[wave * HID + k], h[k], sum);
        // wave32 reduction
        #pragma unroll
        for (int off = 16; off > 0; off >>= 1)
            sum += __shfl_xor(sum, off, 32);
        if (lane == 0) {
            float x = wxr[tok[t]];  // one-hot matvec == column gather
            allH[(size_t)(t + 1) * HID + r] = tanhf(sum + x + bh);
        }
        grid_barrier(barCount, barGen, nblocks);
    }
}

// ---------------------------------------------------------------------------
// Kernel B: outputs = all_h @ W_hy^T via V_WMMA_F32_16X16X4_F32.
// Block tile: M=64 (4 wave M-tiles) x N=128 (8 waves x 16).
// A-block (64 x 1004-padded f32 = 257 KB) staged into LDS with
// GLOBAL_LOAD_ASYNC_TO_LDS_B128 (ASYNCcnt path, no VGPR round-trip),
// conflict-free reads (44 mod 64). Each wave reuses its B fragment across
// 4 accumulators. NT stores for the 524 MB output keep W_hy/all_h resident
// in the 192 MB L2 (HBM roofline ~29 us; compute is far below that).
// ---------------------------------------------------------------------------
#define KPAD 1004

__global__ __launch_bounds__(256) void rnn_out_gemm(
    const float* __restrict__ allH, const float* __restrict__ Why,
    float* __restrict__ out, int M) {
    extern __shared__ float atile[];  // [64][KPAD]

    const int lane = threadIdx.x & 31;
    const int wave = threadIdx.x >> 5;
    const int idx  = lane & 15;   // M (A/D) or N (B/D) index within tile
    const int half = lane >> 4;   // selects K pair {0,1} vs {2,3}
    const int m0   = blockIdx.y * 64;

    // --- Async stage of the A block into LDS -------------------------------
    // ISA flat->LDS mapping truncates upper address bits, so the low 32 bits
    // of the generic shared pointer are the LDS byte address for VDST.
    const unsigned ldsBase = (unsigned)(uintptr_t)atile;
    for (int rr = wave; rr < 64; rr += 8) {
        const int m = m0 + rr;
        if (m < M) {  // wave-uniform branch
            const float* src = allH + (size_t)m * HID;
            const unsigned rowOff = ldsBase + (unsigned)(rr * KPAD) * 4u;
            // 1000 floats = 250 x b128 chunks; both LDS (4016B row pitch)
            // and global (4000B row pitch) sides are 16B aligned.
            for (int c = lane; c < 250; c += 32) {
                asm volatile(
                    "global_load_async_to_lds_b128 %0, %1, off"
                    :
                    : "v"(rowOff + (unsigned)c * 16u), "v"(src + (size_t)c * 4)
                    : "memory");
            }
        } else {  // zero-fill tail rows beyond M
            for (int k = lane; k < HID; k += 32)
                atile[rr * KPAD + k] = 0.f;
        }
    }
    asm volatile("s_wait_asynccnt 0x0" ::: "memory");
    __syncthreads();

    // --- WMMA main loop ----------------------------------------------------
    const int n = blockIdx.x * 128 + wave * 16 + idx;  // 32000 = 250*128 exact
    const float* brow = Why + (size_t)n * HID;         // row n of W_hy == col n of W_hy^T

    v8f acc[4] = {};
    #pragma unroll 4
    for (int k = 0; k < HID; k += 4) {
        const int kb = k + 2 * half;
        v2f b;
        b.x = brow[kb];
        b.y = brow[kb + 1];
        #pragma unroll
        for (int i = 0; i < 4; ++i) {
            v2f a;
            a.x = atile[(i * 16 + idx) * KPAD + kb];
            a.y = atile[(i * 16 + idx) * KPAD + kb + 1];
            acc[i] = __builtin_amdgcn_wmma_f32_16x16x4_f32(
                /*neg_a=*/false, a, /*neg_b=*/false, b,
                /*c_mod=*/(short)0, acc[i],
                /*reuse_a=*/false, /*reuse_b=*/false);
        }
    }

    // --- Epilogue: D layout: VGPR r -> M = r + 8*half, N = idx -------------
    float* outp = out + (size_t)(m0 + 8 * half) * VOC + n;
    if (m0 + 64 <= M) {
        // Fast path (64 of 65 M-blocks): fully branchless NT stores.
        #pragma unroll
        for (int i = 0; i < 4; ++i)
            #pragma unroll
            for (int rr = 0; rr < 8; ++rr)
                __builtin_nontemporal_store(acc[i][rr],
                                            outp + (size_t)(i * 16 + rr) * VOC);
    } else {
        // Tail block: per-row guard.
        #pragma unroll
        for (int i = 0; i < 4; ++i)
            #pragma unroll
            for (int rr = 0; rr < 8; ++rr) {
                const int m = m0 + i * 16 + rr + 8 * half;
                if (m < M)
                    __builtin_nontemporal_store(acc[i][rr],
                                                &out[(size_t)m * VOC + n]);
            }
    }
}

// ---------------------------------------------------------------------------
// Launch. Inputs: 0=token_stream(T,i32) 1=initial_h(1000) 2=W_xh(1000x32000)
// 3=W_hh(1000x1000) 4=W_hy(32000x1000) 5=B_h(1000). Output (T+1)x32000 f32.
// Workspace: all_h ((T+1)*1000 f32, ~16.4 MB) + 2 barrier words.
// ---------------------------------------------------------------------------
extern "C" void kernel_launch(void* const* d_in, const int* in_sizes, int n_in,
                              void* d_out, int out_size, void* d_ws,
                              size_t ws_size, hipStream_t stream) {
    const int*   tok = (const int*)d_in[0];
    const float* h0  = (const float*)d_in[1];
    const float* Wxh = (const float*)d_in[2];
    const float* Whh = (const float*)d_in[3];
    const float* Why = (const float*)d_in[4];
    const float* Bh  = (const float*)d_in[5];
    float* out = (float*)d_out;

    const int T = in_sizes[0];
    const int M = T + 1;

    float* allH = (float*)d_ws;
    size_t allH_bytes = (size_t)M * HID * sizeof(float);
    unsigned* bar = (unsigned*)((char*)d_ws + ((allH_bytes + 255) & ~(size_t)255));

    // Barrier state must start at zero every call (graph-capturable memset).
    hipMemsetAsync(bar, 0, 2 * sizeof(unsigned), stream);

    const unsigned nblocks = 125;  // 125 * 8 rows = 1000 hidden units
    rnn_scan_kernel<<<nblocks, 256, 0, stream>>>(tok, h0, Wxh, Whh, Bh, allH,
                                                 bar, bar + 1, T, nblocks);

    dim3 grid(VOC / 128, (M + 63) / 64);
    size_t smem = (size_t)64 * KPAD * sizeof(float);  // 257 KB (<320 KB/WGP)
    rnn_out_gemm<<<grid, 256, smem, stream>>>(allH, Why, out, M);
}